// GCNStack_87909390615128
// MI455X (gfx1250) — compile-verified
//
#include <hip/hip_runtime.h>

typedef __attribute__((ext_vector_type(2))) float v2f;
typedef __attribute__((ext_vector_type(8))) float v8f;

// ---------------------------------------------------------------------------
// GEMM: out[n x 128] = A[n x 128] @ W[128 x 128] (+ bias), fp32 WMMA 16x16x4.
// One block = 8 waves = one 16-row tile; wave w owns columns [16w,16w+16).
// A tile staged in LDS (8 KB), 32 chained V_WMMA_F32_16X16X4_F32 per wave.
// ---------------------------------------------------------------------------
__global__ __launch_bounds__(256)
void k_gemm_wmma(const float* __restrict__ A, const float* __restrict__ W,
                 const float* __restrict__ bias, float* __restrict__ out)
{
    __shared__ float As[16 * 128];
    const int row0 = blockIdx.x * 16;
    const int tid  = threadIdx.x;
    #pragma unroll
    for (int i = tid; i < 16 * 128; i += 256)
        As[i] = A[(size_t)row0 * 128 + i];
    __syncthreads();

    const int wave    = tid >> 5;      // 0..7 -> column tile
    const int lane    = tid & 31;
    const int hi      = lane >> 4;     // half-wave select
    const int m       = lane & 15;     // row (A) / col (B,D) index
    const int colbase = wave * 16;

    v8f acc = {};
    #pragma unroll
    for (int kb = 0; kb < 128; kb += 4) {
        const int k0 = kb + 2 * hi;    // lanes 16..31 hold K+2 per ISA layout
        v2f a, b;
        a.x = As[m * 128 + k0];
        a.y = As[m * 128 + k0 + 1];
        b.x = W[(size_t)k0       * 128 + colbase + m];
        b.y = W[(size_t)(k0 + 1) * 128 + colbase + m];
        acc = __builtin_amdgcn_wmma_f32_16x16x4_f32(
            /*neg_a=*/false, a, /*neg_b=*/false, b,
            /*c_mod=*/(short)0, acc, /*reuse_a=*/false, /*reuse_b=*/false);
    }
    const float bv = bias ? bias[colbase + m] : 0.0f;
    #pragma unroll
    for (int r = 0; r < 8; ++r) {
        const int row = row0 + r + 8 * hi;   // D layout: VGPR r -> rows r / r+8
        out[(size_t)row * 128 + colbase + m] = acc[r] + bv;
    }
}

// ---------------------------------------------------------------------------
// h = x  (float4 copy)
// ---------------------------------------------------------------------------
__global__ void k_copy4(const float4* __restrict__ src, float4* __restrict__ dst, long n4)
{
    long i = (long)blockIdx.x * blockDim.x + threadIdx.x;
    if (i < n4) dst[i] = src[i];
}

// ---------------------------------------------------------------------------
// Edge embedding + scatter into both endpoints: one wave per edge, lane owns
// 4 of the 128 output dims; ee = attr[e](1x16) @ We(16x128) + be.
// ---------------------------------------------------------------------------
__global__ __launch_bounds__(256)
void k_edge_embed(const float* __restrict__ attr, const int* __restrict__ ei,
                  const float* __restrict__ We, const float* __restrict__ be,
                  float* __restrict__ h, int ne)
{
    long gid = (long)blockIdx.x * 256 + threadIdx.x;
    int  e   = (int)(gid >> 5);
    if (e >= ne) return;
    const int lane = (int)(gid & 31);
    const int d0   = lane * 4;
    const int r    = ei[e];
    const int c    = ei[ne + e];

    float s0 = be[d0], s1 = be[d0 + 1], s2 = be[d0 + 2], s3 = be[d0 + 3];
    #pragma unroll
    for (int k = 0; k < 16; ++k) {
        const float  av = attr[(size_t)e * 16 + k];
        const float4 w  = *(const float4*)(We + k * 128 + d0);
        s0 += av * w.x; s1 += av * w.y; s2 += av * w.z; s3 += av * w.w;
    }
    float* hr = h + (size_t)r * 128 + d0;
    atomicAdd(hr + 0, s0); atomicAdd(hr + 1, s1);
    atomicAdd(hr + 2, s2); atomicAdd(hr + 3, s3);
    float* hc = h + (size_t)c * 128 + d0;
    atomicAdd(hc + 0, s0); atomicAdd(hc + 1, s1);
    atomicAdd(hc + 2, s2); atomicAdd(hc + 3, s3);
}

// ---------------------------------------------------------------------------
// Degree: dinv = rsqrt(1 + indegree(col)).  Computed once, reused 3x.
// ---------------------------------------------------------------------------
__global__ void k_fill1(float* p, int n)
{
    int i = blockIdx.x * blockDim.x + threadIdx.x;
    if (i < n) p[i] = 1.0f;
}
__global__ void k_deg(const int* __restrict__ ei, float* __restrict__ deg, int ne)
{
    int e = blockIdx.x * blockDim.x + threadIdx.x;
    if (e < ne) atomicAdd(&deg[ei[ne + e]], 1.0f);
}
__global__ void k_rsqrt(float* p, int n)
{
    int i = blockIdx.x * blockDim.x + threadIdx.x;
    if (i < n) p[i] = rsqrtf(p[i]);
}

// ---------------------------------------------------------------------------
// agg[i] = t[i] * dinv[i]^2 + bias   (self-loop contribution, norm = 1/deg)
// ---------------------------------------------------------------------------
__global__ __launch_bounds__(256)
void k_self_init(const float* __restrict__ t, const float* __restrict__ dinv,
                 const float* __restrict__ bias, float* __restrict__ agg, int n)
{
    long gid = (long)blockIdx.x * 256 + threadIdx.x;
    int  i   = (int)(gid >> 5);
    if (i >= n) return;
    const int d0 = (int)(gid & 31) * 4;
    const float di = dinv[i];
    const float s  = di * di;
    const float4 v = *(const float4*)(t + (size_t)i * 128 + d0);
    float4 o;
    o.x = v.x * s + bias[d0];
    o.y = v.y * s + bias[d0 + 1];
    o.z = v.z * s + bias[d0 + 2];
    o.w = v.w * s + bias[d0 + 3];
    *(float4*)(agg + (size_t)i * 128 + d0) = o;
}

// ---------------------------------------------------------------------------
// Message pass: agg[col] += t[row] * dinv[row]*dinv[col].  Wave per edge,
// float4 gather + 4 global fp32 atomics per lane.
// ---------------------------------------------------------------------------
__global__ __launch_bounds__(256)
void k_scatter(const float* __restrict__ t, const int* __restrict__ ei,
               const float* __restrict__ dinv, float* __restrict__ agg, int ne)
{
    long gid = (long)blockIdx.x * 256 + threadIdx.x;
    int  e   = (int)(gid >> 5);
    if (e >= ne) return;
    const int d0  = (int)(gid & 31) * 4;
    const int r   = ei[e];
    const int c   = ei[ne + e];
    const float nrm = dinv[r] * dinv[c];
    const float4 v  = *(const float4*)(t + (size_t)r * 128 + d0);
    float* a = agg + (size_t)c * 128 + d0;
    atomicAdd(a + 0, v.x * nrm);
    atomicAdd(a + 1, v.y * nrm);
    atomicAdd(a + 2, v.z * nrm);
    atomicAdd(a + 3, v.w * nrm);
}

// ---------------------------------------------------------------------------
// BatchNorm stats, stage 1: 256 blocks, each reduces its row range into
// per-column partial sum / sumsq (2 threads per column -> LDS combine).
// ---------------------------------------------------------------------------
__global__ __launch_bounds__(256)
void k_bn_partial(const float* __restrict__ x, float* __restrict__ psum,
                  float* __restrict__ psq, int n)
{
    const int d   = threadIdx.x & 127;
    const int sub = threadIdx.x >> 7;            // 0/1: two rows per iteration
    float s = 0.0f, q = 0.0f;
    for (int i = blockIdx.x * 2 + sub; i < n; i += gridDim.x * 2) {
        const float v = x[(size_t)i * 128 + d];
        s += v; q += v * v;
    }
    __shared__ float ls[256], lq[256];
    ls[threadIdx.x] = s; lq[threadIdx.x] = q;
    __syncthreads();
    if (sub == 0) {
        psum[blockIdx.x * 128 + d] = ls[d] + ls[128 + d];
        psq [blockIdx.x * 128 + d] = lq[d] + lq[128 + d];
    }
}

// Stage 2: 128 threads fold the 256 partials, emit mean & rsqrt(var+eps).
__global__ void k_bn_final(const float* __restrict__ psum, const float* __restrict__ psq,
                           float* __restrict__ mean, float* __restrict__ rstd,
                           int nblocks, float invn)
{
    const int d = threadIdx.x;
    float s = 0.0f, q = 0.0f;
    for (int b = 0; b < nblocks; ++b) { s += psum[b * 128 + d]; q += psq[b * 128 + d]; }
    const float mu  = s * invn;
    const float var = q * invn - mu * mu;        // biased variance (training mode)
    mean[d] = mu;
    rstd[d] = rsqrtf(var + 1e-5f);
}

// y = relu((x - mean) * rstd * g + beta)
__global__ void k_bn_relu(const float* __restrict__ x, const float* __restrict__ mean,
                          const float* __restrict__ rstd, const float* __restrict__ g,
                          const float* __restrict__ beta, float* __restrict__ y, long total)
{
    long i = (long)blockIdx.x * blockDim.x + threadIdx.x;
    if (i >= total) return;
    const int d = (int)(i & 127);
    const float v = (x[i] - mean[d]) * rstd[d] * g[d] + beta[d];
    y[i] = v > 0.0f ? v : 0.0f;
}

// ---------------------------------------------------------------------------
extern "C" void kernel_launch(void* const* d_in, const int* in_sizes, int n_in,
                              void* d_out, int out_size, void* d_ws, size_t ws_size,
                              hipStream_t stream)
{
    const float* x    = (const float*)d_in[0];
    const int*   ei   = (const int*)  d_in[1];
    const float* attr = (const float*)d_in[2];
    const float* We   = (const float*)d_in[3];
    const float* be   = (const float*)d_in[4];
    const float* Wc[3] = {(const float*)d_in[5], (const float*)d_in[7], (const float*)d_in[9]};
    const float* bc[3] = {(const float*)d_in[6], (const float*)d_in[8], (const float*)d_in[10]};
    const float* g[2]    = {(const float*)d_in[11], (const float*)d_in[13]};
    const float* beta[2] = {(const float*)d_in[12], (const float*)d_in[14]};
    const float* Wm1 = (const float*)d_in[15]; const float* bm1 = (const float*)d_in[16];
    const float* Wm2 = (const float*)d_in[17]; const float* bm2 = (const float*)d_in[18];
    float* out = (float*)d_out;

    const int n  = in_sizes[0] / 128;   // 100000 (divisible by 16)
    const int ne = in_sizes[1] / 2;     // 1600000

    // Workspace carve-up: h | t | agg | dinv | psum | psq | mean | rstd
    char* ws = (char*)d_ws;
    const size_t nodeBytes = (size_t)n * 128 * sizeof(float);
    float* h   = (float*)(ws);
    float* t   = (float*)(ws + nodeBytes);
    float* agg = (float*)(ws + 2 * nodeBytes);
    char*  aux = ws + 3 * nodeBytes;
    size_t off = ((size_t)n * sizeof(float) + 255) & ~(size_t)255;
    float* dinv = (float*)aux;
    float* psum = (float*)(aux + off);  off += (size_t)256 * 128 * sizeof(float);
    float* psq  = (float*)(aux + off);  off += (size_t)256 * 128 * sizeof(float);
    float* mean = (float*)(aux + off);  off += 128 * sizeof(float);
    float* rstd = (float*)(aux + off);

    const long n4       = (long)n * 32;                 // float4 elements per node buffer
    const long ethreads = (long)ne * 32;                // wave-per-edge thread count
    const int  gemmGrid = (n + 15) / 16;

    // h = x + scatter(edge embeddings)
    k_copy4<<<(int)((n4 + 255) / 256), 256, 0, stream>>>((const float4*)x, (float4*)h, n4);
    k_edge_embed<<<(int)((ethreads + 255) / 256), 256, 0, stream>>>(attr, ei, We, be, h, ne);

    // dinv = rsqrt(deg), computed once for all 3 conv layers
    k_fill1<<<(n + 255) / 256, 256, 0, stream>>>(dinv, n);
    k_deg  <<<(ne + 255) / 256, 256, 0, stream>>>(ei, dinv, ne);
    k_rsqrt<<<(n + 255) / 256, 256, 0, stream>>>(dinv, n);

    const float* cur = h;
    for (int l = 0; l < 3; ++l) {
        k_gemm_wmma<<<gemmGrid, 256, 0, stream>>>(cur, Wc[l], nullptr, t);
        k_self_init<<<(int)((n4 * 8 / 8 + 255) / 256), 256, 0, stream>>>(t, dinv, bc[l], agg, n);
        k_scatter<<<(int)((ethreads + 255) / 256), 256, 0, stream>>>(t, ei, dinv, agg, ne);
        if (l < 2) {
            k_bn_partial<<<256, 256, 0, stream>>>(agg, psum, psq, n);
            k_bn_final<<<1, 128, 0, stream>>>(psum, psq, mean, rstd, 256, 1.0f / (float)n);
            const long total = (long)n * 128;
            k_bn_relu<<<(int)((total + 255) / 256), 256, 0, stream>>>(agg, mean, rstd,
                                                                      g[l], beta[l], h, total);
            cur = h;
        }
    }

    // post_mp: two biased GEMMs
    k_gemm_wmma<<<gemmGrid, 256, 0, stream>>>(agg, Wm1, bm1, h);
    k_gemm_wmma<<<gemmGrid, 256, 0, stream>>>(h, Wm2, bm2, out);
}